// MyAttention_67714454389225
// MI455X (gfx1250) — compile-verified
//
#include <hip/hip_runtime.h>
#include <math.h>

typedef __attribute__((ext_vector_type(2))) float v2f;
typedef __attribute__((ext_vector_type(8))) float v8f;

#define EE   128     // embedding dim
#define HH   128     // hidden dim
#define BB   32      // batch
#define SS   512     // sequence length
#define CC   5       // classes
#define G4   512     // 4*H
#define HP   132     // padded h row stride (dwords) -> bank-conflict free
#define GP   516     // padded gates row stride (dwords)

__device__ __forceinline__ float sigf(float v) {
    return 1.0f / (1.0f + __expf(-v));
}

// -------------------------------------------------------------------------
// Kernel 1: Xp[s*32+b][0..511] = embed[x[b,s]] @ W_ih^T + (b_ih + b_hh)
// One wave per 16x16 output tile, K-loop of V_WMMA_F32_16X16X4_F32.
// Row index r = s*32 + b so each timestep slab Xp[t][:][:] is contiguous.
// -------------------------------------------------------------------------
__global__ __launch_bounds__(256)
void xproj_kernel(const int* __restrict__ x,
                  const float* __restrict__ embed,
                  const float* __restrict__ Wih,
                  const float* __restrict__ bih,
                  const float* __restrict__ bhh,
                  float* __restrict__ Xp)
{
    const int lane = threadIdx.x & 31;
    const int wave = threadIdx.x >> 5;
    const int gw   = blockIdx.x * 8 + wave;   // global wave id = tile id
    const int rt   = gw >> 5;                 // 1024 row tiles
    const int ct   = gw & 31;                 // 32 col tiles
    const int r0   = rt * 16;
    const int n0   = ct * 16;
    const int m    = lane & 15;               // M (A) / N (B,C) index
    const int g2   = (lane >> 4) << 1;        // K sub-offset: 0 or 2
    const int g8   = (lane >> 4) << 3;        // C row offset: 0 or 8

    // Gather the A-matrix row for this lane (embedding lookup)
    const int rr  = r0 + m;
    const int s   = rr >> 5;                  // B == 32
    const int b   = rr & 31;
    const int tok = x[b * SS + s];
    const float* arow = embed + (size_t)tok * EE;
    const float* brow = Wih + (size_t)(n0 + m) * EE;  // B[k][n] = Wih[n][k]

    v8f c = {};
#pragma unroll
    for (int kk = 0; kk < EE / 4; ++kk) {
        const int k0 = kk * 4 + g2;
        v2f a  = *(const v2f*)(arow + k0);
        v2f bf = *(const v2f*)(brow + k0);
        c = __builtin_amdgcn_wmma_f32_16x16x4_f32(false, a, false, bf,
                                                  (short)0, c, false, false);
    }

    const float bias = bih[n0 + m] + bhh[n0 + m];
#pragma unroll
    for (int i = 0; i < 8; ++i) {
        Xp[(size_t)(r0 + i + g8) * G4 + n0 + m] = c[i] + bias;
    }
}

// -------------------------------------------------------------------------
// Kernel 2: sequential LSTM. One workgroup, 32 waves (wave32).
// Wave w owns gate-column tile [16w,16w+16) AND batch b=w for the
// elementwise/score phase. W_hh^T fragment resident in VGPRs for all steps.
// -------------------------------------------------------------------------
__global__ __launch_bounds__(1024)
void lstm_kernel(const float* __restrict__ Xp,
                 const float* __restrict__ Whh,
                 const float* __restrict__ myw,
                 const float* __restrict__ linW,
                 const float* __restrict__ linb,
                 float* __restrict__ out)
{
    extern __shared__ float smem[];
    float* hbuf = smem;             // [32][HP] hidden state
    float* gbuf = smem + BB * HP;   // [32][GP] recurrent gate pre-activations

    const int tid  = threadIdx.x;
    const int lane = tid & 31;
    const int wave = tid >> 5;            // 0..31
    const int m    = lane & 15;
    const int g2   = (lane >> 4) << 1;
    const int g8   = (lane >> 4) << 3;
    const int n0   = wave * 16;           // gate column tile

    // --- B-matrix (W_hh^T) fragment: 32 x v2f = 64 VGPRs, constant all steps
    v2f barr[32];
    {
        const float* wrow = Whh + (size_t)(n0 + m) * HH;
#pragma unroll
        for (int kk = 0; kk < 32; ++kk)
            barr[kk] = *(const v2f*)(wrow + kk * 4 + g2);
    }

    // zero initial hidden state
    for (int i = tid; i < BB * HP; i += 1024) hbuf[i] = 0.0f;

    // elementwise ownership: batch eb = wave, columns ehh..ehh+3
    const int eb  = wave;
    const int ehh = lane * 4;
    float cell[4] = {0.f, 0.f, 0.f, 0.f};
    float selh[4] = {0.f, 0.f, 0.f, 0.f};
    float best = -3.0e38f;
    float mw[4];
#pragma unroll
    for (int j = 0; j < 4; ++j) mw[j] = myw[ehh + j];

    __syncthreads();

    for (int t = 0; t < SS; ++t) {
        // ---- GEMM phase: gates_h(n0-tile) = h @ Whh^T, both batch-row tiles
        v8f c0 = {}, c1 = {};
#pragma unroll
        for (int kk = 0; kk < 32; ++kk) {
            const int k0 = kk * 4 + g2;
            v2f a0 = *(const v2f*)(hbuf + m * HP + k0);
            v2f a1 = *(const v2f*)(hbuf + (16 + m) * HP + k0);
            c0 = __builtin_amdgcn_wmma_f32_16x16x4_f32(false, a0, false, barr[kk],
                                                       (short)0, c0, false, false);
            c1 = __builtin_amdgcn_wmma_f32_16x16x4_f32(false, a1, false, barr[kk],
                                                       (short)0, c1, false, false);
        }
#pragma unroll
        for (int i = 0; i < 8; ++i) {
            gbuf[(i + g8) * GP + n0 + m]      = c0[i];
            gbuf[(16 + i + g8) * GP + n0 + m] = c1[i];
        }
        __syncthreads();

        // ---- Elementwise phase: LSTM cell update + score for batch eb
        const float* xrow = Xp + ((size_t)t * BB + eb) * G4;
        if (t + 1 < SS)  // hint next step's input-projection slab toward L0/L2
            __builtin_prefetch(Xp + ((size_t)(t + 1) * BB + eb) * G4 + ehh, 0, 0);

        const float* gb = gbuf + eb * GP;
        float hv[4];
        float part = 0.0f;
#pragma unroll
        for (int j = 0; j < 4; ++j) {
            float iv = sigf (gb[           ehh + j] + xrow[           ehh + j]);
            float fv = sigf (gb[    HH +   ehh + j] + xrow[    HH +   ehh + j]);
            float gv = tanhf(gb[2 * HH +   ehh + j] + xrow[2 * HH +   ehh + j]);
            float ov = sigf (gb[3 * HH +   ehh + j] + xrow[3 * HH +   ehh + j]);
            cell[j] = fv * cell[j] + iv * gv;
            hv[j]   = ov * tanhf(cell[j]);
            hbuf[eb * HP + ehh + j] = hv[j];
            part += hv[j] * mw[j];
        }
        // wave32 reduction of h . myw (all lanes end with the total)
#pragma unroll
        for (int off = 16; off >= 1; off >>= 1)
            part += __shfl_xor(part, off, 32);
        const float score = tanhf(part);   // exp() is monotone -> skip for argmax
        if (score > best) {                // strict > keeps FIRST max (matches ref)
            best = score;
#pragma unroll
            for (int j = 0; j < 4; ++j) selh[j] = hv[j];
        }
        __syncthreads();
    }

    // ---- Final linear: out[b] = sel @ linW^T + linb, wave w == batch w
#pragma unroll
    for (int cc = 0; cc < CC; ++cc) {
        float p = 0.0f;
#pragma unroll
        for (int j = 0; j < 4; ++j) p += selh[j] * linW[cc * HH + ehh + j];
#pragma unroll
        for (int off = 16; off >= 1; off >>= 1)
            p += __shfl_xor(p, off, 32);
        if (lane == 0) out[eb * CC + cc] = p + linb[cc];
    }
}

// -------------------------------------------------------------------------
extern "C" void kernel_launch(void* const* d_in, const int* in_sizes, int n_in,
                              void* d_out, int out_size, void* d_ws, size_t ws_size,
                              hipStream_t stream) {
    const int*   x     = (const int*)  d_in[0];
    const float* embed = (const float*)d_in[1];
    const float* Wih   = (const float*)d_in[2];
    const float* Whh   = (const float*)d_in[3];
    const float* bih   = (const float*)d_in[4];
    const float* bhh   = (const float*)d_in[5];
    const float* myw   = (const float*)d_in[6];
    const float* linW  = (const float*)d_in[7];
    const float* linb  = (const float*)d_in[8];
    float* out = (float*)d_out;
    float* Xp  = (float*)d_ws;          // [S*B][4H] = 32 MB

    // Kernel 1: 32768 tiles, 8 waves (256 threads) per block
    xproj_kernel<<<4096, 256, 0, stream>>>(x, embed, Wih, bih, bhh, Xp);

    // Kernel 2: single persistent workgroup, 83 KB dynamic LDS (>64K default)
    const size_t smem = (size_t)(BB * HP + BB * GP) * sizeof(float);
    (void)hipFuncSetAttribute((const void*)lstm_kernel,
                              hipFuncAttributeMaxDynamicSharedMemorySize,
                              (int)smem);
    lstm_kernel<<<1, 1024, smem, stream>>>(Xp, Whh, myw, linW, linb, out);
}